// MaxpoolMatchLay_1580547973065
// MI455X (gfx1250) — compile-verified
//
#include <hip/hip_runtime.h>
#include <hip/hip_bf16.h>

typedef __attribute__((ext_vector_type(2))) float v2f;
typedef __attribute__((ext_vector_type(8))) float v8f;

#define LDA 260          // padded row stride (floats) to spread LDS banks
#define MP_EPS 1e-8f

// cont_repres:       [16, 256, 256] f32
// other_cont_repres: [16, 256, 256] f32
// weight:            [64, 256]      f32
// out:               [16, 256, 128] f32  (first 64 = max over j, last 64 = mean over j)
__global__ __launch_bounds__(256) void mp_cosine_match_kernel(
    const float* __restrict__ A,
    const float* __restrict__ B,
    const float* __restrict__ W,
    float* __restrict__ out)
{
    __shared__ __align__(16) float wRow[256];
    __shared__ __align__(16) float waTile[64 * LDA];   // weighted A rows (i-block)
    __shared__ __align__(16) float wbTile[32 * LDA];   // weighted B rows (j-step)
    __shared__ float naS[64];
    __shared__ float nbS[32];
    __shared__ float partMax[64][2];
    __shared__ float partSum[64][2];

    const int tid  = threadIdx.x;
    const int lane = tid & 31;
    const int wave = tid >> 5;       // 0..7
    const int isub = wave >> 1;      // 0..3 : which 16-row strip of the 64-row i-block
    const int jsub = wave & 1;       // 0..1 : which 16-col strip of the 32-col j-step

    const int ib = blockIdx.x;       // 0..3  i-block
    const int m  = blockIdx.y;       // 0..63 perspective
    const int b  = blockIdx.z;       // 0..15 batch
    const int i0 = ib * 64;

    // ---- stage weight row ----
    wRow[tid] = W[m * 256 + tid];
    __syncthreads();

    // ---- stage weighted A tile: 64 rows x 256 cols (float4 path) ----
    {
        const int r  = tid >> 2;             // 0..63
        const int c0 = tid & 3;              // float4 column phase
        const float* arow = A + ((size_t)b * 256 + (i0 + r)) * 256;
        float4* dst = (float4*)&waTile[r * LDA];
        #pragma unroll 4
        for (int c4 = c0; c4 < 64; c4 += 4) {
            float4 av = ((const float4*)arow)[c4];
            float4 wv = ((const float4*)wRow)[c4];
            float4 p;
            p.x = av.x * wv.x; p.y = av.y * wv.y;
            p.z = av.z * wv.z; p.w = av.w * wv.w;
            dst[c4] = p;
        }
    }
    __syncthreads();

    // ---- norms of weighted A rows ----
    if (tid < 64) {
        const float* row = &waTile[tid * LDA];
        float s = 0.f;
        #pragma unroll 4
        for (int d = 0; d < 256; ++d) { float v = row[d]; s += v * v; }
        naS[tid] = sqrtf(s);
    }

    float rmax[8], rsum[8];
    #pragma unroll
    for (int v = 0; v < 8; ++v) { rmax[v] = -1.0e30f; rsum[v] = 0.f; }

    // WMMA fragment addressing (f32 16x16x4):
    //   A 16x4: lane L -> row = L&15, VGPR0/1 hold K = (L>>4)*2, +1
    //   B 4x16: lane L -> col = L&15, VGPR0/1 hold K = (L>>4)*2, +1
    const int fr = lane & 15;
    const int kh = (lane >> 4) << 1;         // 0 or 2
    const float* aPtr = &waTile[(isub * 16 + fr) * LDA + kh];
    const float* bPtrBase = &wbTile[(jsub * 16 + fr) * LDA + kh];

    // ---- loop over j in steps of 32 columns ----
    for (int js = 0; js < 8; ++js) {
        __syncthreads();                      // protect wbTile reuse

        // stage weighted B tile: 32 rows x 256 cols
        {
            const int r  = tid >> 3;          // 0..31
            const int c0 = tid & 7;
            const float* brow = B + ((size_t)b * 256 + (js * 32 + r)) * 256;
            float4* dst = (float4*)&wbTile[r * LDA];
            #pragma unroll 4
            for (int c4 = c0; c4 < 64; c4 += 8) {
                float4 bv = ((const float4*)brow)[c4];
                float4 wv = ((const float4*)wRow)[c4];
                float4 p;
                p.x = bv.x * wv.x; p.y = bv.y * wv.y;
                p.z = bv.z * wv.z; p.w = bv.w * wv.w;
                dst[c4] = p;
            }
        }
        __syncthreads();
        if (tid < 32) {
            const float* row = &wbTile[tid * LDA];
            float s = 0.f;
            #pragma unroll 4
            for (int d = 0; d < 256; ++d) { float v = row[d]; s += v * v; }
            nbS[tid] = sqrtf(s);
        }
        __syncthreads();

        // ---- 16x16 output tile, K=256 via v_wmma_f32_16x16x4_f32 ----
        v8f c = {0.f, 0.f, 0.f, 0.f, 0.f, 0.f, 0.f, 0.f};
        #pragma unroll 8
        for (int k0 = 0; k0 < 256; k0 += 4) {
            v2f af = *(const v2f*)(aPtr + k0);
            v2f bf = *(const v2f*)(bPtrBase + k0);
            c = __builtin_amdgcn_wmma_f32_16x16x4_f32(
                    /*neg_a=*/false, af, /*neg_b=*/false, bf,
                    /*c_mod=*/(short)0, c, /*reuse_a=*/false, /*reuse_b=*/false);
        }

        // ---- cosine normalize + fold into running max/mean ----
        // C/D layout: lane L, VGPR v -> M = v + (L>=16 ? 8 : 0), N = L&15
        const float nbv   = nbS[jsub * 16 + (lane & 15)];
        const int   mBase = isub * 16 + ((lane >= 16) ? 8 : 0);
        #pragma unroll
        for (int v = 0; v < 8; ++v) {
            float denom = fmaxf(naS[mBase + v] * nbv, MP_EPS);
            float simi  = c[v] / denom;
            rmax[v] = fmaxf(rmax[v], simi);
            rsum[v] += simi;
        }
    }

    // ---- reduce over the 16 lanes holding one output row (stay in 16-lane halves) ----
    #pragma unroll
    for (int v = 0; v < 8; ++v) {
        #pragma unroll
        for (int off = 1; off <= 8; off <<= 1) {
            rmax[v] = fmaxf(rmax[v], __shfl_xor(rmax[v], off, 32));
            rsum[v] += __shfl_xor(rsum[v], off, 32);
        }
    }
    if (lane == 0 || lane == 16) {
        const int mBase = isub * 16 + ((lane == 16) ? 8 : 0);
        #pragma unroll
        for (int v = 0; v < 8; ++v) {
            partMax[mBase + v][jsub] = rmax[v];
            partSum[mBase + v][jsub] = rsum[v];
        }
    }
    __syncthreads();

    // ---- combine the two j-waves, write out[b, i, m] and out[b, i, 64+m] ----
    if (tid < 64) {
        float mx = fmaxf(partMax[tid][0], partMax[tid][1]);
        float mn = (partSum[tid][0] + partSum[tid][1]) * (1.0f / 256.0f);
        float* o = out + ((size_t)b * 256 + (i0 + tid)) * 128;
        o[m]      = mx;
        o[64 + m] = mn;
    }
}

extern "C" void kernel_launch(void* const* d_in, const int* in_sizes, int n_in,
                              void* d_out, int out_size, void* d_ws, size_t ws_size,
                              hipStream_t stream) {
    (void)in_sizes; (void)n_in; (void)d_ws; (void)ws_size; (void)out_size;
    const float* A = (const float*)d_in[0];   // cont_repres       [16,256,256]
    const float* B = (const float*)d_in[1];   // other_cont_repres [16,256,256]
    const float* W = (const float*)d_in[2];   // weight            [64,256]
    float* out = (float*)d_out;               // [16,256,128]

    dim3 grid(4, 64, 16);                     // (i-block, perspective, batch)
    dim3 block(256);
    mp_cosine_match_kernel<<<grid, block, 0, stream>>>(A, B, W, out);
}